// Pacemaker_42107859370236
// MI455X (gfx1250) — compile-verified
//
#include <hip/hip_runtime.h>
#include <stdint.h>

// ---------------- constants from the reference ----------------
#define DT      0.001f                     // dt = 1/1000
#define NS      0.018973665961010276f      // 0.6 * sqrt(0.001)
#define PBASE   0.003f                     // 3.0 * dt
#define CHUNK   64                         // timesteps per chunk
#define DEPTH   8                          // async pipeline stages per wave
#define NTHR    256                        // threads per block (8 waves)

typedef float v4f __attribute__((ext_vector_type(4)));

// One OU + spike step on 4 neurons.
__device__ __forceinline__ v4f step4(v4f& x, v4f n4, v4f u4) {
    v4f s;
#pragma unroll
    for (int i = 0; i < 4; ++i) {
        float xv = x[i];
        xv = xv - xv * DT + n4[i] * NS;     // OU update (ref op order)
        x[i] = xv;
        float m = 0.1f * tanhf(xv);
        float p = PBASE * (1.0f + m);
        p = fminf(fmaxf(p, 0.0f), 1.0f);
        s[i] = (u4[i] < p) ? 1.0f : 0.0f;
    }
    return s;
}

// -------- kernel 1: per-chunk OU contribution with zero initial state -------
// Default-RT loads: this noise is re-read by pm_spikes, let L2 keep what it can.
__global__ __launch_bounds__(NTHR)
void pm_partials(const float* __restrict__ noise, float* __restrict__ partials, int N) {
    const int c  = blockIdx.y;
    const int vg = blockIdx.x * NTHR + threadIdx.x;   // 4-neuron group
    const int n0 = vg * 4;
    const float* pn = noise + (size_t)c * CHUNK * N + n0;
    v4f x = (v4f)0.0f;
#pragma unroll 8
    for (int t = 0; t < CHUNK; ++t) {
        v4f n4 = *(const v4f*)pn;
#pragma unroll
        for (int i = 0; i < 4; ++i)
            x[i] = x[i] - x[i] * DT + n4[i] * NS;
        pn += N;
    }
    *(v4f*)(partials + (size_t)c * N + n0) = x;
}

// -------- kernel 2: serial carry scan over chunks (tiny) --------------------
__global__ __launch_bounds__(NTHR)
void pm_scan(const float* __restrict__ state, const float* __restrict__ partials,
             float* __restrict__ carries, int N, int C) {
    const int n = blockIdx.x * NTHR + threadIdx.x;
    if (n >= N) return;
    float A = 1.0f;                         // (per-step decay)^CHUNK, f32-consistent
#pragma unroll
    for (int i = 0; i < CHUNK; ++i) A = A - A * DT;
    float X = state[n];
    for (int c = 0; c < C; ++c) {
        carries[(size_t)c * N + n] = X;
        X = A * X + partials[(size_t)c * N + n];
    }
}

// -------- kernel 3: spike generation with async global->LDS pipeline --------
// Each wave owns DEPTH stages x 1KB of LDS (32 lanes x (16B noise + 16B uniform)).
// Streaming data is loaded non-temporally (never re-read; keep L2 clean).
__global__ __launch_bounds__(NTHR)
void pm_spikes(const float* __restrict__ noise, const float* __restrict__ uniform,
               const float* __restrict__ carries, float* __restrict__ out, int N) {
    __shared__ v4f smem[(NTHR / 32) * DEPTH * 64];    // 64 KB total

    const int tid  = threadIdx.x;
    const int wave = tid >> 5;
    const int lane = tid & 31;
    const int c    = blockIdx.y;
    const int vg   = blockIdx.x * NTHR + tid;
    const int n0   = vg * 4;

    const size_t baseT = (size_t)c * CHUNK * N;
    const float* qn = noise   + baseT + n0;           // per-lane 16B slice
    const float* qu = uniform + baseT + n0;
    float*       po = out     + baseT + n0;

    // LDS byte addresses (low 32 bits of a generic LDS pointer == LDS offset)
    const int idx_wave = wave * (DEPTH * 64);         // in v4f units
    const uint32_t lds_wave   = (uint32_t)(uintptr_t)&smem[idx_wave];
    const uint32_t lds_lane_n = lds_wave + (uint32_t)lane * 16u;
    const uint32_t lds_lane_u = lds_lane_n + 512u;

    // Prologue: fill all DEPTH stages (2 async loads each -> ASYNCcnt = 16)
#pragma unroll
    for (int s = 0; s < DEPTH; ++s) {
        uint32_t ln = lds_lane_n + (uint32_t)s * 1024u;
        uint32_t lu = lds_lane_u + (uint32_t)s * 1024u;
        asm volatile(
            "global_load_async_to_lds_b128 %0, %2, off th:TH_LOAD_NT\n\t"
            "global_load_async_to_lds_b128 %1, %3, off th:TH_LOAD_NT"
            :: "v"(ln), "v"(lu), "v"(qn), "v"(qu)
            : "memory");
        qn += N; qu += N;
    }

    v4f x = *(const v4f*)(carries + (size_t)c * N + n0);

    int t = 0;
    for (; t < CHUNK - DEPTH; ++t) {
        // wait until the oldest stage's 2 loads completed: asynccnt <= 2*(DEPTH-1)
        asm volatile("s_wait_asynccnt 14" ::: "memory");
        const int slot = t & (DEPTH - 1);
        v4f n4 = smem[idx_wave + slot * 64 + lane];
        v4f u4 = smem[idx_wave + slot * 64 + 32 + lane];
        v4f s4 = step4(x, n4, u4);
        __builtin_nontemporal_store(s4, (v4f*)po);
        po += N;
        // refill the slot we just consumed; guard LDS WAR with s_wait_dscnt 0
        uint32_t ln = lds_lane_n + (uint32_t)slot * 1024u;
        uint32_t lu = lds_lane_u + (uint32_t)slot * 1024u;
        asm volatile(
            "s_wait_dscnt 0\n\t"
            "global_load_async_to_lds_b128 %0, %2, off th:TH_LOAD_NT\n\t"
            "global_load_async_to_lds_b128 %1, %3, off th:TH_LOAD_NT"
            :: "v"(ln), "v"(lu), "v"(qn), "v"(qu)
            : "memory");
        qn += N; qu += N;
    }

    asm volatile("s_wait_asynccnt 0" ::: "memory");
    for (; t < CHUNK; ++t) {
        const int slot = t & (DEPTH - 1);
        v4f n4 = smem[idx_wave + slot * 64 + lane];
        v4f u4 = smem[idx_wave + slot * 64 + 32 + lane];
        v4f s4 = step4(x, n4, u4);
        __builtin_nontemporal_store(s4, (v4f*)po);
        po += N;
    }
}

// -------- fallback: monolithic sequential scan (used if ws too small) -------
__global__ __launch_bounds__(NTHR)
void pm_fallback(const float* __restrict__ state, const float* __restrict__ noise,
                 const float* __restrict__ uniform, float* __restrict__ out,
                 int N, int T) {
    const int vg = blockIdx.x * NTHR + threadIdx.x;
    const int n0 = vg * 4;
    if (n0 >= N) return;
    v4f x = *(const v4f*)(state + n0);
    const float* pn = noise + n0;
    const float* pu = uniform + n0;
    float*       po = out + n0;
    for (int t = 0; t < T; ++t) {
        v4f n4 = *(const v4f*)pn;
        v4f u4 = *(const v4f*)pu;
        v4f s4 = step4(x, n4, u4);
        __builtin_nontemporal_store(s4, (v4f*)po);
        pn += N; pu += N; po += N;
    }
}

extern "C" void kernel_launch(void* const* d_in, const int* in_sizes, int n_in,
                              void* d_out, int out_size, void* d_ws, size_t ws_size,
                              hipStream_t stream) {
    // dict order: T (scalar, device), state [N], noise [T,N], uniform [T,N]
    const float* state   = (const float*)d_in[1];
    const float* noise   = (const float*)d_in[2];
    const float* uniform = (const float*)d_in[3];
    float* out = (float*)d_out;

    const int N = in_sizes[1];
    const int T = in_sizes[2] / N;
    const int C = T / CHUNK;

    const size_t need = 2 * (size_t)C * (size_t)N * sizeof(float);
    const bool chunked = (T % CHUNK == 0) && (N % (4 * NTHR) == 0) &&
                         (C >= 1) && (ws_size >= need);

    if (!chunked) {
        const int groups = (N + 3) / 4;
        pm_fallback<<<(groups + NTHR - 1) / NTHR, NTHR, 0, stream>>>(
            state, noise, uniform, out, N, T);
        return;
    }

    float* partials = (float*)d_ws;                    // [C, N]
    float* carries  = partials + (size_t)C * N;        // [C, N]

    dim3 grid(N / (4 * NTHR), C);
    pm_partials<<<grid, NTHR, 0, stream>>>(noise, partials, N);
    pm_scan<<<(N + NTHR - 1) / NTHR, NTHR, 0, stream>>>(state, partials, carries, N, C);
    pm_spikes<<<grid, NTHR, 0, stream>>>(noise, uniform, carries, out, N);
}